// GNNBlock_35613868819062
// MI455X (gfx1250) — compile-verified
//
#include <hip/hip_runtime.h>
#include <hip/hip_bf16.h>

typedef __attribute__((ext_vector_type(2))) float v2f;
typedef __attribute__((ext_vector_type(8))) float v8f;

#define IN_CH  48
#define OUT_CH 48
#define BN_EPS 1e-5f

// ---------------------------------------------------------------------------
// 1) Workspace init: agg = 0, deg = 1 (self loop), stats = 0
// ---------------------------------------------------------------------------
__global__ void init_ws_kernel(float* __restrict__ agg, float* __restrict__ deg,
                               float* __restrict__ stats, int n_nodes) {
    int stride = gridDim.x * blockDim.x;
    int total  = n_nodes * OUT_CH;
    for (int t = blockIdx.x * blockDim.x + threadIdx.x; t < total; t += stride) {
        agg[t] = 0.0f;
        if (t < n_nodes)    deg[t]   = 1.0f;   // self loop contributes 1 to degree
        if (t < 2 * OUT_CH) stats[t] = 0.0f;
    }
}

// ---------------------------------------------------------------------------
// 2) Degree at destination (self loops pre-counted by init)
// ---------------------------------------------------------------------------
__global__ void degree_kernel(const int* __restrict__ dst, float* __restrict__ deg, int E) {
    int e = blockIdx.x * blockDim.x + threadIdx.x;
    if (e < E) unsafeAtomicAdd(&deg[dst[e]], 1.0f);
}

// ---------------------------------------------------------------------------
// 3) dinv = rsqrt(deg), in place
// ---------------------------------------------------------------------------
__global__ void rsqrt_kernel(float* __restrict__ d, int n) {
    int i = blockIdx.x * blockDim.x + threadIdx.x;
    if (i < n) d[i] = rsqrtf(d[i]);
}

// ---------------------------------------------------------------------------
// 4) h = x @ W via V_WMMA_F32_16X16X4_F32.
//    blockDim = 256 (8 waves); each wave owns a 16-row strip of x and produces
//    three 16x16 f32 tiles (OUT_CH = 48). K = 48 -> 12 WMMA steps per tile.
//    W is repacked in LDS as K-pairs (v2f) so each B fragment is one aligned
//    8-byte DS load straight into an even-aligned VGPR pair (no v_mov fixup).
//    A frag (16x4): lane(l16,half): M=l16, VGPR0=K(k0+2*half), VGPR1=K+1.
//    B frag (4x16): lane(l16,half): N=l16, VGPR0=K(k0+2*half), VGPR1=K+1.
//    C/D: lanes 0-15 hold M=v, lanes 16-31 hold M=v+8, N=l16 per VGPR v.
// ---------------------------------------------------------------------------
__global__ void __launch_bounds__(256)
gemm_wmma_kernel(const float* __restrict__ x, const float* __restrict__ W,
                 float* __restrict__ h, int n_nodes, int n_rowblocks) {
    __shared__ v2f   sWp[(IN_CH / 2) * OUT_CH];  // [kk][n] = (W[2kk][n], W[2kk+1][n])
    __shared__ float sX[8][16 * 50];             // per-wave 16x48 tile, stride 50 (pad)

    const int tid  = threadIdx.x;
    const int wv   = tid >> 5;
    const int lane = tid & 31;
    const int half = lane >> 4;   // 0: lanes 0-15, 1: lanes 16-31
    const int l16  = lane & 15;

    // Cooperative load + repack of W into LDS (block-shared); W is tiny (9 KB)
    for (int i = tid; i < (IN_CH / 2) * OUT_CH; i += 256) {
        int kk = i / OUT_CH;
        int n  = i - kk * OUT_CH;
        v2f p;
        p.x = W[(2 * kk) * OUT_CH + n];
        p.y = W[(2 * kk + 1) * OUT_CH + n];
        sWp[i] = p;
    }

    const int rb      = blockIdx.x * 8 + wv;   // uniform within a wave
    const int rowBase = rb * 16;

    // Per-wave load of its 16x48 X tile (tail rows clamped; stores guarded later)
    if (rb < n_rowblocks) {
        for (int i = lane; i < 16 * IN_CH; i += 32) {
            int r  = i / IN_CH;
            int c  = i - r * IN_CH;
            int gr = rowBase + r;
            if (gr > n_nodes - 1) gr = n_nodes - 1;
            sX[wv][r * 50 + c] = x[(size_t)gr * IN_CH + c];
        }
    }
    __syncthreads();

    if (rb >= n_rowblocks) return;   // whole-wave uniform exit

    v8f c0 = {}, c1 = {}, c2 = {};
    const float* xs = &sX[wv][0];

#pragma unroll
    for (int k0 = 0; k0 < IN_CH; k0 += 4) {
        const int k  = k0 + half * 2;          // even -> 8B-aligned in sX
        const int kk = (k0 >> 1) + half;       // K-pair index into sWp
        v2f a  = *(const v2f*)(xs + l16 * 50 + k);
        v2f b0 = sWp[kk * OUT_CH + l16];
        v2f b1 = sWp[kk * OUT_CH + 16 + l16];
        v2f b2 = sWp[kk * OUT_CH + 32 + l16];
        c0 = __builtin_amdgcn_wmma_f32_16x16x4_f32(false, a, false, b0, (short)0, c0, false, false);
        c1 = __builtin_amdgcn_wmma_f32_16x16x4_f32(false, a, false, b1, (short)0, c1, false, false);
        c2 = __builtin_amdgcn_wmma_f32_16x16x4_f32(false, a, false, b2, (short)0, c2, false, false);
    }

#pragma unroll
    for (int v = 0; v < 8; ++v) {
        int m   = v + half * 8;
        int row = rowBase + m;
        if (row < n_nodes) {
            float* hr = h + (size_t)row * OUT_CH;
            hr[l16]      = c0[v];
            hr[16 + l16] = c1[v];
            hr[32 + l16] = c2[v];
        }
    }
}

// ---------------------------------------------------------------------------
// 5) Edge gather/scale/scatter. 12 float4 chunks per (edge|self-loop).
//    h and agg are L2-resident (19.2 MB each << 192 MB), so gathers and the
//    fp32 atomics run at L2 speed.
// ---------------------------------------------------------------------------
__global__ void scatter_kernel(const int* __restrict__ src, const int* __restrict__ dst,
                               const float* __restrict__ dinv, const float* __restrict__ h,
                               float* __restrict__ agg, int E, int n_nodes) {
    long long t     = (long long)blockIdx.x * blockDim.x + threadIdx.x;
    long long total = (long long)(E + n_nodes) * 12;
    if (t >= total) return;
    int e = (int)(t / 12);
    int q = (int)(t - (long long)e * 12);

    int s, d; float nrm;
    if (e < E) {
        s = src[e]; d = dst[e];
        nrm = dinv[s] * dinv[d];
    } else {
        s = d = e - E;                 // self loop
        float di = dinv[s];
        nrm = di * di;
    }

    const float4 v = *((const float4*)(h + (size_t)s * OUT_CH) + q);
    float* ap = agg + (size_t)d * OUT_CH + q * 4;
    unsafeAtomicAdd(ap + 0, v.x * nrm);
    unsafeAtomicAdd(ap + 1, v.y * nrm);
    unsafeAtomicAdd(ap + 2, v.z * nrm);
    unsafeAtomicAdd(ap + 3, v.w * nrm);
}

// ---------------------------------------------------------------------------
// 6) BN statistics: per-channel sum & sumsq of (agg + b).
//    block = (48, 8); LDS tree over the 8 row-lanes, then atomics into stats.
// ---------------------------------------------------------------------------
__global__ void __launch_bounds__(384)
stats_kernel(const float* __restrict__ agg, const float* __restrict__ b,
             float* __restrict__ stats, int n_nodes) {
    __shared__ float s1[8][OUT_CH];
    __shared__ float s2[8][OUT_CH];
    const int c  = threadIdx.x;   // channel 0..47
    const int ty = threadIdx.y;   // 0..7

    const float bc = b[c];
    float sum = 0.0f, sq = 0.0f;
    for (int n = blockIdx.x * 8 + ty; n < n_nodes; n += gridDim.x * 8) {
        float v = agg[(size_t)n * OUT_CH + c] + bc;
        sum += v;
        sq  += v * v;
    }
    s1[ty][c] = sum;
    s2[ty][c] = sq;
    __syncthreads();
    if (ty == 0) {
        float a1 = 0.0f, a2 = 0.0f;
#pragma unroll
        for (int j = 0; j < 8; ++j) { a1 += s1[j][c]; a2 += s2[j][c]; }
        unsafeAtomicAdd(&stats[c], a1);
        unsafeAtomicAdd(&stats[OUT_CH + c], a2);
    }
}

// ---------------------------------------------------------------------------
// 7) Normalize + affine + ReLU -> out
// ---------------------------------------------------------------------------
__global__ void final_kernel(const float* __restrict__ agg, const float* __restrict__ b,
                             const float* __restrict__ gamma, const float* __restrict__ beta,
                             const float* __restrict__ stats, float* __restrict__ out,
                             int n_nodes) {
    const int   total  = n_nodes * OUT_CH;
    const float invN   = 1.0f / (float)n_nodes;
    const int   stride = gridDim.x * blockDim.x;
    for (int t = blockIdx.x * blockDim.x + threadIdx.x; t < total; t += stride) {
        int   c    = t % OUT_CH;
        float mean = stats[c] * invN;
        float var  = stats[OUT_CH + c] * invN - mean * mean;
        float v    = agg[t] + b[c];
        float y    = (v - mean) * rsqrtf(var + BN_EPS) * gamma[c] + beta[c];
        out[t] = fmaxf(y, 0.0f);
    }
}

// ---------------------------------------------------------------------------
extern "C" void kernel_launch(void* const* d_in, const int* in_sizes, int n_in,
                              void* d_out, int out_size, void* d_ws, size_t ws_size,
                              hipStream_t stream) {
    const float* x     = (const float*)d_in[0];
    const int*   ei    = (const int*)d_in[1];
    const float* W     = (const float*)d_in[2];
    const float* b     = (const float*)d_in[3];
    const float* gamma = (const float*)d_in[4];
    const float* beta  = (const float*)d_in[5];
    float*       out   = (float*)d_out;

    const int N = in_sizes[0] / IN_CH;
    const int E = in_sizes[1] / 2;
    const int* src = ei;       // edge_index[0]
    const int* dst = ei + E;   // edge_index[1]

    // Workspace layout: h[N*48] | agg[N*48] | dinv[N] | stats[96]
    float* h     = (float*)d_ws;
    float* agg   = h + (size_t)N * OUT_CH;
    float* dinv  = agg + (size_t)N * OUT_CH;
    float* stats = dinv + N;

    const int T = 256;

    {   // init
        int total  = N * OUT_CH;
        int blocks = (total + T - 1) / T;
        init_ws_kernel<<<blocks, T, 0, stream>>>(agg, dinv, stats, N);
    }
    {   // degree
        int blocks = (E + T - 1) / T;
        degree_kernel<<<blocks, T, 0, stream>>>(dst, dinv, E);
    }
    {   // rsqrt
        int blocks = (N + T - 1) / T;
        rsqrt_kernel<<<blocks, T, 0, stream>>>(dinv, N);
    }
    {   // GEMM (WMMA f32)
        int nrb    = (N + 15) / 16;
        int blocks = (nrb + 7) / 8;
        gemm_wmma_kernel<<<blocks, 256, 0, stream>>>(x, W, h, N, nrb);
    }
    {   // edge scatter
        long long total = (long long)(E + N) * 12;
        int blocks = (int)((total + T - 1) / T);
        scatter_kernel<<<blocks, T, 0, stream>>>(src, dst, dinv, h, agg, E, N);
    }
    {   // BN stats
        dim3 blk(OUT_CH, 8);
        stats_kernel<<<1024, blk, 0, stream>>>(agg, b, stats, N);
    }
    {   // finalize
        int total  = N * OUT_CH;
        int blocks = (total + T - 1) / T;
        final_kernel<<<blocks, T, 0, stream>>>(agg, b, gamma, beta, stats, out, N);
    }
}